// GATConv_14894946583453
// MI455X (gfx1250) — compile-verified
//
#include <hip/hip_runtime.h>
#include <math.h>

typedef __attribute__((ext_vector_type(2))) float v2f;
typedef __attribute__((ext_vector_type(8))) float v8f;

#define D_DIM 256
#define HF 256
#define DE_DIM 16
#define NEG_SLOPE 0.2f
#define KS (D_DIM + 4)   // LDS row stride (floats): +4 pad -> conflict-free b64 reads

// monotonic float <-> uint mapping for atomicMax-based segment max
__device__ __forceinline__ unsigned fkey(float f) {
  unsigned b = __float_as_uint(f);
  return (b & 0x80000000u) ? ~b : (b | 0x80000000u);
}
__device__ __forceinline__ float funkey(unsigned u) {
  unsigned b = (u & 0x80000000u) ? (u & 0x7FFFFFFFu) : ~u;
  return __uint_as_float(b);
}

// ---------------------------------------------------------------- init
__global__ void gat_init_kernel(unsigned* __restrict__ emax_u,
                                float* __restrict__ denom, int n4) {
  int i = blockIdx.x * blockDim.x + threadIdx.x;
  if (i < n4) { emax_u[i] = 0u; denom[i] = 0.0f; }
}

// ---------------------------------------------------------------- dual GEMM (WMMA f32)
// src_fc = feat @ Wsrc^T        -> workspace
// out    = feat @ Wdst^T + bias -> d_out (residual base, later atomically accumulated)
// B staged in LDS (shared by all 8 waves); A prefetched at distance 2, B at distance 1.
__global__ void __launch_bounds__(256)
gat_gemm_dual_wmma(const float* __restrict__ feat,
                   const float* __restrict__ Wsrc,
                   const float* __restrict__ Wdst,
                   const float* __restrict__ bias,
                   float* __restrict__ src_fc,
                   float* __restrict__ out,
                   int N) {
  __shared__ float smem[2 * 64 * KS];   // [matrix][col 0..63][k 0..255 (+pad)]

  const int tid   = threadIdx.x;
  const int wave  = tid >> 5;
  const int lane  = tid & 31;
  const int half  = lane >> 4;     // 0: lanes 0-15 (K=0,1), 1: lanes 16-31 (K=2,3)
  const int lan16 = lane & 15;
  const int m0 = blockIdx.x * 128 + wave * 16;   // 8 waves * 16 rows per block
  const int n0 = blockIdx.y * 64;                // 4 col-tiles of 16 per wave

  // ---- cooperative LDS stage of both weight slices (coalesced float4) ----
  {
    const int total4 = 2 * 64 * (D_DIM / 4);     // 8192 float4
    for (int i = tid; i < total4; i += 256) {
      int m   = i >> 12;          // 0 = Wsrc, 1 = Wdst
      int rem = i & 4095;
      int c   = rem >> 6;
      int k4  = rem & 63;
      const float* W = m ? Wdst : Wsrc;
      float4 v = *(const float4*)(W + (size_t)(n0 + c) * D_DIM + k4 * 4);
      *(float4*)(smem + (size_t)(m * 64 + c) * KS + k4 * 4) = v;
    }
  }
  __syncthreads();

  // A row per lane, clamped (WMMA needs EXEC all-1s; guard only the stores)
  int arow = m0 + lan16;
  if (arow > N - 1) arow = N - 1;
  const float* Aptr = feat + (size_t)arow * D_DIM + half * 2;

  // LDS fragment base pointers: B[K][N=col] lives at smem[(m*64+col)*KS + k]
  const float* Bs[4];
  const float* Bd[4];
#pragma unroll
  for (int t = 0; t < 4; ++t) {
    int c = t * 16 + lan16;
    Bs[t] = smem + (size_t)c * KS + half * 2;
    Bd[t] = smem + (size_t)(64 + c) * KS + half * 2;
  }

  v8f accS[4], accD[4];
#pragma unroll
  for (int t = 0; t < 4; ++t) { accS[t] = (v8f)0.0f; accD[t] = (v8f)0.0f; }

  // ---- software-pipelined K loop ----
  // A: distance-2 rotation (aA consumed now, aB in flight-1, a_fut issued now)
  // B: distance-1 from LDS
  v2f aA = *(const v2f*)(Aptr);        // k = 0
  v2f aB = *(const v2f*)(Aptr + 4);    // k = 4
  v2f bs_cur[4], bd_cur[4];
#pragma unroll
  for (int t = 0; t < 4; ++t) {
    bs_cur[t] = *(const v2f*)(Bs[t]);
    bd_cur[t] = *(const v2f*)(Bd[t]);
  }

  for (int k = 0; k < D_DIM; k += 4) {
    const int kn  = (k + 4 < D_DIM) ? (k + 4) : k;   // distance-1 (LDS B)
    const int kn2 = (k + 8 < D_DIM) ? (k + 8) : kn;  // distance-2 (global A)
    v2f a_fut = *(const v2f*)(Aptr + kn2);
    v2f bs_nxt[4], bd_nxt[4];
#pragma unroll
    for (int t = 0; t < 4; ++t) {
      bs_nxt[t] = *(const v2f*)(Bs[t] + kn);
      bd_nxt[t] = *(const v2f*)(Bd[t] + kn);
    }
#pragma unroll
    for (int t = 0; t < 4; ++t) {
      accS[t] = __builtin_amdgcn_wmma_f32_16x16x4_f32(
          false, aA, false, bs_cur[t], (short)0, accS[t], false, false);
      accD[t] = __builtin_amdgcn_wmma_f32_16x16x4_f32(
          false, aA, false, bd_cur[t], (short)0, accD[t], false, false);
    }
    aA = aB;
    aB = a_fut;
#pragma unroll
    for (int t = 0; t < 4; ++t) { bs_cur[t] = bs_nxt[t]; bd_cur[t] = bd_nxt[t]; }
  }

  // C/D layout: VGPR r -> row m0+r (lanes 0-15) / m0+8+r (lanes 16-31)
#pragma unroll
  for (int t = 0; t < 4; ++t) {
    int col = n0 + t * 16 + lan16;
    float bv = bias[col];
#pragma unroll
    for (int r = 0; r < 8; ++r) {
      int row = m0 + r + half * 8;
      if (row < N) {
        src_fc[(size_t)row * HF + col] = accS[t][r];
        out[(size_t)row * HF + col]    = accD[t][r] + bv;
      }
    }
  }
}

// ---------------------------------------------------------------- node attention dots
__global__ void __launch_bounds__(256)
gat_attn_node_kernel(const float* __restrict__ feat,
                     const float* __restrict__ Was,
                     const float* __restrict__ Wad,
                     float* __restrict__ attn_src,
                     float* __restrict__ attn_dst, int N) {
  int wave = threadIdx.x >> 5, lane = threadIdx.x & 31;
  int n = blockIdx.x * 8 + wave;
  if (n >= N) return;
  float ss[4] = {0.f, 0.f, 0.f, 0.f};
  float sd[4] = {0.f, 0.f, 0.f, 0.f};
  const float* f = feat + (size_t)n * D_DIM;
  for (int j = lane; j < D_DIM; j += 32) {
    float fv = f[j];
#pragma unroll
    for (int h = 0; h < 4; ++h) {
      ss[h] += fv * Was[h * D_DIM + j];
      sd[h] += fv * Wad[h * D_DIM + j];
    }
  }
#pragma unroll
  for (int h = 0; h < 4; ++h) {
    for (int off = 16; off > 0; off >>= 1) {
      ss[h] += __shfl_xor(ss[h], off, 32);
      sd[h] += __shfl_xor(sd[h], off, 32);
    }
  }
  if (lane == 0) {
#pragma unroll
    for (int h = 0; h < 4; ++h) {
      attn_src[n * 4 + h] = ss[h];
      attn_dst[n * 4 + h] = sd[h];
    }
  }
}

// ---------------------------------------------------------------- edge logits + segment max
__global__ void gat_edge_logit_kernel(const float* __restrict__ fe,
                                      const int* __restrict__ src,
                                      const int* __restrict__ dst,
                                      const float* __restrict__ Wae,
                                      const float* __restrict__ attn_src,
                                      const float* __restrict__ attn_dst,
                                      float* __restrict__ e_arr,
                                      unsigned* __restrict__ emax_u, int E) {
  int e = blockIdx.x * blockDim.x + threadIdx.x;
  if (e >= E) return;
  int s = src[e], d = dst[e];
  const float* fep = fe + (size_t)e * DE_DIM;
  float fcache[DE_DIM];
#pragma unroll
  for (int j = 0; j < DE_DIM; ++j) fcache[j] = fep[j];
#pragma unroll
  for (int h = 0; h < 4; ++h) {
    float ae = 0.f;
#pragma unroll
    for (int j = 0; j < DE_DIM; ++j) ae += fcache[j] * Wae[h * DE_DIM + j];
    float v = attn_src[s * 4 + h] + attn_dst[d * 4 + h] + ae;
    v = (v >= 0.f) ? v : NEG_SLOPE * v;   // leaky relu
    e_arr[(size_t)e * 4 + h] = v;
    atomicMax(&emax_u[d * 4 + h], fkey(v));
  }
}

// ---------------------------------------------------------------- exp + segment sum
__global__ void gat_edge_exp_kernel(const int* __restrict__ dst,
                                    const unsigned* __restrict__ emax_u,
                                    float* __restrict__ e_arr,
                                    float* __restrict__ denom, int E) {
  int e = blockIdx.x * blockDim.x + threadIdx.x;
  if (e >= E) return;
  int d = dst[e];
#pragma unroll
  for (int h = 0; h < 4; ++h) {
    float ex = expf(e_arr[(size_t)e * 4 + h] - funkey(emax_u[d * 4 + h]));
    e_arr[(size_t)e * 4 + h] = ex;
    __hip_atomic_fetch_add(&denom[d * 4 + h], ex,
                           __ATOMIC_RELAXED, __HIP_MEMORY_SCOPE_AGENT);
  }
}

// ---------------------------------------------------------------- weighted gather/scatter
// one wave per edge: gather src_fc[src] (1KB, coalesced), scale per head, fp32-atomic add
__global__ void __launch_bounds__(256)
gat_aggregate_kernel(const int* __restrict__ src,
                     const int* __restrict__ dst,
                     const float* __restrict__ e_arr,
                     const float* __restrict__ denom,
                     const float* __restrict__ src_fc,
                     float* __restrict__ out, int E) {
  int wave = threadIdx.x >> 5, lane = threadIdx.x & 31;
  int e = blockIdx.x * 8 + wave;
  if (e >= E) return;
  int s = src[e], d = dst[e];
  float a4[4];
#pragma unroll
  for (int h = 0; h < 4; ++h)
    a4[h] = e_arr[(size_t)e * 4 + h] / denom[d * 4 + h];
  const float* fp = src_fc + (size_t)s * HF;
  float* op = out + (size_t)d * HF;
#pragma unroll
  for (int j = 0; j < 8; ++j) {      // 8 x 32 lanes = 256 elems; head = j/2
    int i = j * 32 + lane;
    float v = fp[i] * a4[j >> 1];
    __hip_atomic_fetch_add(&op[i], v,
                           __ATOMIC_RELAXED, __HIP_MEMORY_SCOPE_AGENT);
  }
}

// ================================================================ launch
extern "C" void kernel_launch(void* const* d_in, const int* in_sizes, int n_in,
                              void* d_out, int out_size, void* d_ws, size_t ws_size,
                              hipStream_t stream) {
  const float* feat        = (const float*)d_in[0];
  const float* feat_edge   = (const float*)d_in[1];
  const int*   src         = (const int*)d_in[2];
  const int*   dst         = (const int*)d_in[3];
  const float* W_src       = (const float*)d_in[4];
  const float* W_dst       = (const float*)d_in[5];
  const float* b_dst       = (const float*)d_in[6];
  const float* W_attn_src  = (const float*)d_in[7];
  const float* W_attn_dst  = (const float*)d_in[8];
  const float* W_attn_edge = (const float*)d_in[9];
  float* out = (float*)d_out;

  const int N = in_sizes[0] / D_DIM;
  const int E = in_sizes[2];

  // workspace layout (floats): src_fc[N*256] | e_arr[E*4] | attn_src[N*4] |
  //                            attn_dst[N*4] | emax_u[N*4] | denom[N*4]
  float*    ws       = (float*)d_ws;
  float*    src_fc   = ws;
  float*    e_arr    = src_fc + (size_t)N * HF;
  float*    attn_src = e_arr + (size_t)E * 4;
  float*    attn_dst = attn_src + (size_t)N * 4;
  unsigned* emax_u   = (unsigned*)(attn_dst + (size_t)N * 4);
  float*    denom    = (float*)(emax_u + (size_t)N * 4);

  // 1. zero segment-max keys and denominators (graph-replay safe init)
  int n4 = N * 4;
  gat_init_kernel<<<(n4 + 255) / 256, 256, 0, stream>>>(emax_u, denom, n4);

  // 2. WMMA dual GEMM: src_fc -> ws, dst_fc + bias -> d_out (residual base)
  dim3 gg((N + 127) / 128, HF / 64);
  gat_gemm_dual_wmma<<<gg, 256, 0, stream>>>(feat, W_src, W_dst, b_dst,
                                             src_fc, out, N);

  // 3. per-node attention dots
  gat_attn_node_kernel<<<(N + 7) / 8, 256, 0, stream>>>(
      feat, W_attn_src, W_attn_dst, attn_src, attn_dst, N);

  // 4. edge logits (leaky relu) + segment max via keyed atomicMax
  gat_edge_logit_kernel<<<(E + 255) / 256, 256, 0, stream>>>(
      feat_edge, src, dst, W_attn_edge, attn_src, attn_dst, e_arr, emax_u, E);

  // 5. exp(e - max) + segment sum via fp32 atomics
  gat_edge_exp_kernel<<<(E + 255) / 256, 256, 0, stream>>>(
      dst, emax_u, e_arr, denom, E);

  // 6. normalized weighted gather/scatter into d_out (adds on top of dst_fc)
  gat_aggregate_kernel<<<(E + 7) / 8, 256, 0, stream>>>(
      src, dst, e_arr, denom, src_fc, out, E);
}